// VectorQuantizerEMA_74483322847343
// MI455X (gfx1250) — compile-verified
//
#include <hip/hip_runtime.h>
#include <hip/hip_bf16.h>

typedef __attribute__((ext_vector_type(2))) float v2f;
typedef __attribute__((ext_vector_type(4))) float v4f;
typedef __attribute__((ext_vector_type(8))) float v8f;

#define K_EMB   1024
#define D_EMB   64
#define N_ROWS  65536      // 16 * 64 * 64
#define HW      4096       // 64 * 64
#define DECAYF  0.99f
#define EPSF    1e-5f

// workspace layout (floats)
#define WS_E2H   0          // 1024 : 0.5*||e_k||^2
#define WS_CNT   1024       // 1024 : encoding counts
#define WS_DW    2048       // 65536: dw[ch*1024 + k]
#define WS_LOSS  67584      // 1    : sum of squared diffs
#define WS_BPK   67588      // 65536: WMMA-layout packed codebook (16B aligned)

// output layout (floats, reference tuple order)
#define OUT_Q    0          // 4194304 : quantized_st  [b,c,h,w]
#define OUT_LOSS 4194304    // 1       : commitment loss
#define OUT_IDX  4194305    // 65536   : indices (as float)
#define OUT_EMB  4259841    // 65536   : new_embedding [d,K]
#define OUT_CS   4325377    // 1024    : new_cluster_size
#define OUT_EA   4326401    // 65536   : new_embed_avg [d,K]

__global__ __launch_bounds__(256)
void vq_e2h_kernel(const float* __restrict__ E, float* __restrict__ ws) {
    int k = blockIdx.x * blockDim.x + threadIdx.x;
    if (k < K_EMB) {
        float s = 0.0f;
        #pragma unroll 8
        for (int d = 0; d < D_EMB; ++d) {
            float e = E[d * K_EMB + k];
            s += e * e;
        }
        ws[WS_E2H + k] = 0.5f * s;
    }
}

// Permute codebook into WMMA B-operand lane layout: per (chunk,i2,lane) a
// contiguous float4 = rows {8*i2+2*hl, +1, +4, +5} at column chunk*16+l16,
// covering K-steps 2*i2 (xy) and 2*i2+1 (zw).
__global__ __launch_bounds__(256)
void vq_bpack_kernel(const float* __restrict__ E, float* __restrict__ ws) {
    int tid   = blockIdx.x * blockDim.x + threadIdx.x;   // 64*8*32 = 16384
    int lane  = tid & 31;
    int i2    = (tid >> 5) & 7;
    int chunk = tid >> 8;
    int hl    = lane >> 4;
    int l16   = lane & 15;
    int kcol  = chunk * 16 + l16;
    int r0    = 8 * i2 + 2 * hl;
    v4f v;
    v.x = E[(r0    ) * K_EMB + kcol];
    v.y = E[(r0 + 1) * K_EMB + kcol];
    v.z = E[(r0 + 4) * K_EMB + kcol];
    v.w = E[(r0 + 5) * K_EMB + kcol];
    ((v4f*)(ws + WS_BPK))[tid] = v;
}

// One wave handles a 32-row tile (two 16-row WMMA subtiles sharing each B
// fragment). Scores vs all 1024 codes via V_WMMA_F32_16X16X4_F32.
__global__ __launch_bounds__(128)
void vq_assign_kernel(const float* __restrict__ X, const float* __restrict__ E,
                      float* ws, float* out) {
    __shared__ int sK[4 * 32];

    const int lane = threadIdx.x & 31;
    const int wave = threadIdx.x >> 5;
    const int tile = blockIdx.x * 4 + wave;
    const int hl   = lane >> 4;
    const int l16  = lane & 15;

    const int n0 = tile * 32 + l16;          // subtile 0 row for this lane
    const int n1 = n0 + 16;                  // subtile 1 row
    const int bi0 = n0 >> 12, sp0 = n0 & (HW - 1);
    const int bi1 = n1 >> 12, sp1 = n1 & (HW - 1);
    const float* base0 = X + (size_t)bi0 * D_EMB * HW + sp0;
    const float* base1 = X + (size_t)bi1 * D_EMB * HW + sp1;

    // A operands (ISA 16x4 fp32 layout): lanes 0-15 K={4i,4i+1}, 16-31 K={4i+2,4i+3}
    v2f A0[16], A1[16];
    #pragma unroll
    for (int i = 0; i < 16; ++i) {
        int c0 = 4 * i + 2 * hl;
        A0[i].x = base0[(size_t)c0 * HW];
        A0[i].y = base0[(size_t)(c0 + 1) * HW];
        A1[i].x = base1[(size_t)c0 * HW];
        A1[i].y = base1[(size_t)(c0 + 1) * HW];
    }

    float bV0[8], bV1[8];
    int   bI0[8], bI1[8];
    #pragma unroll
    for (int j = 0; j < 8; ++j) {
        bV0[j] = 3.402823466e38f; bI0[j] = 0;
        bV1[j] = 3.402823466e38f; bI1[j] = 0;
    }

    const v4f* __restrict__ Bpk = (const v4f*)(ws + WS_BPK);

    for (int chunk = 0; chunk < 64; ++chunk) {
        const int kcol   = chunk * 16 + l16;
        const float e2hv = ws[WS_E2H + kcol];

        v8f c0 = {0.f,0.f,0.f,0.f,0.f,0.f,0.f,0.f};
        v8f c1 = {0.f,0.f,0.f,0.f,0.f,0.f,0.f,0.f};
        #pragma unroll
        for (int i2 = 0; i2 < 8; ++i2) {
            v4f bv = Bpk[(chunk * 8 + i2) * 32 + lane];   // global_load_b128
            v2f blo; blo.x = bv.x; blo.y = bv.y;          // K-step 2*i2
            v2f bhi; bhi.x = bv.z; bhi.y = bv.w;          // K-step 2*i2+1
            c0 = __builtin_amdgcn_wmma_f32_16x16x4_f32(
                     false, A0[2*i2    ], false, blo, (short)0, c0, false, false);
            c1 = __builtin_amdgcn_wmma_f32_16x16x4_f32(
                     false, A1[2*i2    ], false, blo, (short)0, c1, false, false);
            c0 = __builtin_amdgcn_wmma_f32_16x16x4_f32(
                     false, A0[2*i2 + 1], false, bhi, (short)0, c0, false, false);
            c1 = __builtin_amdgcn_wmma_f32_16x16x4_f32(
                     false, A1[2*i2 + 1], false, bhi, (short)0, c1, false, false);
        }
        #pragma unroll
        for (int j = 0; j < 8; ++j) {
            float v0 = e2hv - c0[j];   // score/2 = ||e||^2/2 - x.e
            float v1 = e2hv - c1[j];
            if (v0 < bV0[j]) { bV0[j] = v0; bI0[j] = kcol; }
            if (v1 < bV1[j]) { bV1[j] = v1; bI1[j] = kcol; }
        }
    }

    // Argmin reduce across the 16 lanes of each half (both subtiles)
    #pragma unroll
    for (int m = 1; m <= 8; m <<= 1) {
        #pragma unroll
        for (int j = 0; j < 8; ++j) {
            float ov = __shfl_xor(bV0[j], m, 32);
            int   oi = __shfl_xor(bI0[j], m, 32);
            if (ov < bV0[j] || (ov == bV0[j] && oi < bI0[j])) { bV0[j] = ov; bI0[j] = oi; }
            ov = __shfl_xor(bV1[j], m, 32);
            oi = __shfl_xor(bI1[j], m, 32);
            if (ov < bV1[j] || (ov == bV1[j] && oi < bI1[j])) { bV1[j] = ov; bI1[j] = oi; }
        }
    }
    if (l16 == 0) {
        #pragma unroll
        for (int j = 0; j < 8; ++j) {
            sK[wave * 32      + j + 8 * hl] = bI0[j];
            sK[wave * 32 + 16 + j + 8 * hl] = bI1[j];
        }
    }
    __syncthreads();
    const int myK0 = sK[wave * 32      + l16];
    const int myK1 = sK[wave * 32 + 16 + l16];

    if (hl == 0) {
        atomicAdd(&ws[WS_CNT + myK0], 1.0f);
        atomicAdd(&ws[WS_CNT + myK1], 1.0f);
        out[OUT_IDX + n0] = (float)myK0;
        out[OUT_IDX + n1] = (float)myK1;
    }

    // Quantized gather/store, dw atomics, loss; reuses A registers
    float* q0 = out + OUT_Q + (size_t)bi0 * D_EMB * HW + sp0;
    float* q1 = out + OUT_Q + (size_t)bi1 * D_EMB * HW + sp1;
    float lsum = 0.0f;
    #pragma unroll
    for (int i = 0; i < 16; ++i) {
        int c0 = 4 * i + 2 * hl;   // halves cover disjoint channels per row
        {
            float x0 = A0[i].x, x1 = A0[i].y;
            float e0 = E[(size_t)c0 * K_EMB + myK0];
            float e1 = E[(size_t)(c0 + 1) * K_EMB + myK0];
            q0[(size_t)c0 * HW]       = e0;
            q0[(size_t)(c0 + 1) * HW] = e1;
            atomicAdd(&ws[WS_DW + c0 * K_EMB + myK0], x0);
            atomicAdd(&ws[WS_DW + (c0 + 1) * K_EMB + myK0], x1);
            float d0 = e0 - x0, d1 = e1 - x1;
            lsum += d0 * d0 + d1 * d1;
        }
        {
            float x0 = A1[i].x, x1 = A1[i].y;
            float e0 = E[(size_t)c0 * K_EMB + myK1];
            float e1 = E[(size_t)(c0 + 1) * K_EMB + myK1];
            q1[(size_t)c0 * HW]       = e0;
            q1[(size_t)(c0 + 1) * HW] = e1;
            atomicAdd(&ws[WS_DW + c0 * K_EMB + myK1], x0);
            atomicAdd(&ws[WS_DW + (c0 + 1) * K_EMB + myK1], x1);
            float d0 = e0 - x0, d1 = e1 - x1;
            lsum += d0 * d0 + d1 * d1;
        }
    }
    #pragma unroll
    for (int m = 16; m >= 1; m >>= 1) lsum += __shfl_xor(lsum, m, 32);
    if (lane == 0) atomicAdd(&ws[WS_LOSS], lsum);
}

__global__ __launch_bounds__(1024)
void vq_finalize_kernel(const float* __restrict__ cs_in, const float* __restrict__ ea_in,
                        const float* __restrict__ ws, float* __restrict__ out) {
    __shared__ float red[1024];
    const int k = threadIdx.x;

    float ncs = cs_in[k] * DECAYF + (1.0f - DECAYF) * ws[WS_CNT + k];
    out[OUT_CS + k] = ncs;
    red[k] = ncs;
    __syncthreads();
    for (int s = 512; s > 0; s >>= 1) {
        if (k < s) red[k] += red[k + s];
        __syncthreads();
    }
    const float nsum = red[0];

    const float cs  = (ncs + EPSF) / (nsum + K_EMB * EPSF) * nsum;
    const float inv = 1.0f / cs;
    for (int d = 0; d < D_EMB; ++d) {
        float nea = ea_in[d * K_EMB + k] * DECAYF
                  + (1.0f - DECAYF) * ws[WS_DW + d * K_EMB + k];
        out[OUT_EA  + d * K_EMB + k] = nea;
        out[OUT_EMB + d * K_EMB + k] = nea * inv;
    }
    if (k == 0) out[OUT_LOSS] = 0.25f * ws[WS_LOSS] / 4194304.0f;
}

extern "C" void kernel_launch(void* const* d_in, const int* in_sizes, int n_in,
                              void* d_out, int out_size, void* d_ws, size_t ws_size,
                              hipStream_t stream) {
    (void)in_sizes; (void)n_in; (void)out_size; (void)ws_size;
    const float* X  = (const float*)d_in[0];   // inputs       [16,64,64,64]
    const float* E  = (const float*)d_in[1];   // embedding    [64,1024]
    const float* CS = (const float*)d_in[2];   // cluster_size [1024]
    const float* EA = (const float*)d_in[3];   // embed_avg    [64,1024]
    float* out = (float*)d_out;
    float* ws  = (float*)d_ws;

    // zero counts + dw + loss accumulator (capture-safe async memset)
    hipMemsetAsync(ws + WS_CNT, 0,
                   (size_t)(K_EMB + K_EMB * D_EMB + 1) * sizeof(float), stream);

    vq_e2h_kernel<<<K_EMB / 256, 256, 0, stream>>>(E, ws);
    vq_bpack_kernel<<<16384 / 256, 256, 0, stream>>>(E, ws);
    vq_assign_kernel<<<N_ROWS / 32 / 4, 128, 0, stream>>>(X, E, ws, out);
    vq_finalize_kernel<<<1, 1024, 0, stream>>>(CS, EA, ws, out);
}